// PostProcessor_45930380263939
// MI455X (gfx1250) — compile-verified
//
#include <hip/hip_runtime.h>
#include <hip/hip_bf16.h>
#include <cfloat>
#include <cstdint>
#include <cstddef>

// Problem constants (match reference)
constexpr int B_  = 128;
constexpr int R_  = 6320;
constexpr int N_  = 80;
constexpr int CR_ = 51;    // relation classes
constexpr int CO_ = 151;   // object classes

constexpr int SORT_SIZE = 8192;   // next pow2 >= R_
constexpr int SORT_NT   = 1024;

// ---------------------------------------------------------------------------
// CDNA5 async global->LDS (gfx1250). Guarded: falls back to load+ds_store.
// #warning markers surface in stderr so we can confirm which path compiled.
// ---------------------------------------------------------------------------
#if defined(__gfx1250__) && \
    __has_builtin(__builtin_amdgcn_global_load_async_to_lds_b32) && \
    __has_builtin(__builtin_amdgcn_s_wait_asynccnt)
#define USE_ASYNC_LDS 1
#warning "CDNA5 path: using global_load_async_to_lds_b32 + s_wait_asynccnt"
#else
#define USE_ASYNC_LDS 0
#if defined(__gfx1250__)
#warning "CDNA5 path: async-to-LDS builtins NOT available, using ds_store fallback"
#endif
#endif

// Non-temporal store: final outputs are write-once; keep them from evicting
// the L2-resident rel_logits working set (165MB vs 192MB L2).
template <typename T>
__device__ __forceinline__ void nt_store(T* p, T v) {
    __builtin_nontemporal_store(v, p);
}

// ---------------------------------------------------------------------------
// Kernel 1: object softmax scores. One wave32 per (b, n) row of 151 logits.
//   out_pred[row]   = argmax_{c>=1} softmax   (as float, label in [1,150])
//   out_scores[row] = max_{c>=1} softmax
// ---------------------------------------------------------------------------
__global__ void obj_score_kernel(const float* __restrict__ obj_logits,
                                 float* __restrict__ out_pred,
                                 float* __restrict__ out_scores,
                                 float* __restrict__ ws_obj)
{
    const int wave = (int)((blockIdx.x * blockDim.x + threadIdx.x) >> 5);
    const int lane = threadIdx.x & 31;
    if (wave >= B_ * N_) return;

    const float* base = obj_logits + (size_t)wave * CO_;

    float v[5];
    float m  = -FLT_MAX;          // max over ALL classes (softmax stability)
    float bl = -FLT_MAX;          // best logit over classes >= 1
    int   bi = 0x7FFFFFFF;
#pragma unroll
    for (int k = 0; k < 5; ++k) {
        const int c = lane + 32 * k;
        const float x = (c < CO_) ? base[c] : -FLT_MAX;
        v[k] = x;
        m = fmaxf(m, x);
        if (c >= 1 && c < CO_ && x > bl) { bl = x; bi = c; }
    }
    for (int o = 16; o > 0; o >>= 1) m = fmaxf(m, __shfl_xor(m, o, 32));
    for (int o = 16; o > 0; o >>= 1) {
        const float ol = __shfl_xor(bl, o, 32);
        const int   oi = __shfl_xor(bi, o, 32);
        if (ol > bl || (ol == bl && oi < bi)) { bl = ol; bi = oi; }
    }
    float s = 0.f;
#pragma unroll
    for (int k = 0; k < 5; ++k) {
        const int c = lane + 32 * k;
        if (c < CO_) s += __expf(v[k] - m);
    }
    for (int o = 16; o > 0; o >>= 1) s += __shfl_xor(s, o, 32);

    if (lane == 0) {
        const float sc = __expf(bl - m) / s;
        nt_store(&out_pred[wave],   (float)bi);
        nt_store(&out_scores[wave], sc);
        ws_obj[wave] = sc;                    // re-read by kernel 2: keep cached
    }
}

// ---------------------------------------------------------------------------
// Kernel 2: relation softmax score + triple score (sort key). One wave per
// relation row of 51 logits. key = rel_score * s0 * s1;
// val = (rel_class << 16) | r  (r < 6320 fits 13 bits, class < 51 fits 6).
// rel_logits read with default temporal hint so it stays hot in L2 for k4.
// ---------------------------------------------------------------------------
__global__ void rel_key_kernel(const float* __restrict__ rel_logits,
                               const int*   __restrict__ pair_idx,
                               const float* __restrict__ ws_obj,
                               float*       __restrict__ ws_keys,
                               unsigned*    __restrict__ ws_vals)
{
    const int wave = (int)((blockIdx.x * blockDim.x + threadIdx.x) >> 5);
    const int lane = threadIdx.x & 31;
    if (wave >= B_ * R_) return;

    const float* base = rel_logits + (size_t)wave * CR_;
    const float l0 = base[lane];
    const float l1 = (lane < CR_ - 32) ? base[lane + 32] : -FLT_MAX;

    float m = fmaxf(l0, l1);
    for (int o = 16; o > 0; o >>= 1) m = fmaxf(m, __shfl_xor(m, o, 32));

    float s = __expf(l0 - m) + ((lane < CR_ - 32) ? __expf(l1 - m) : 0.f);
    for (int o = 16; o > 0; o >>= 1) s += __shfl_xor(s, o, 32);

    // argmax over classes 1..50, first-occurrence tie-break
    float bl = (lane >= 1) ? l0 : -FLT_MAX;
    int   bi = (lane >= 1) ? lane : 0x7FFFFFFF;
    if (lane < CR_ - 32 && l1 > bl) { bl = l1; bi = lane + 32; }
    for (int o = 16; o > 0; o >>= 1) {
        const float ol = __shfl_xor(bl, o, 32);
        const int   oi = __shfl_xor(bi, o, 32);
        if (ol > bl || (ol == bl && oi < bi)) { bl = ol; bi = oi; }
    }

    if (lane == 0) {
        const int b = wave / R_;
        const int r = wave - b * R_;
        const int2 pr = ((const int2*)pair_idx)[wave];
        const float s0 = ws_obj[b * N_ + pr.x];
        const float s1 = ws_obj[b * N_ + pr.y];
        const float rel_score = __expf(bl - m) / s;
        ws_keys[wave] = rel_score * s0 * s1;          // re-read by kernel 3
        ws_vals[wave] = (unsigned)r | ((unsigned)bi << 16);
    }
}

// ---------------------------------------------------------------------------
// Kernel 3: per-image bitonic sort of 8192 (key,val) pairs in 64KB LDS.
// Order: key descending, index ascending on ties (stable argsort(-key)).
// Emits sorted index (ws), sorted labels (out), gathered pairs (out).
// ---------------------------------------------------------------------------
__global__ __launch_bounds__(SORT_NT)
void sort_kernel(const float*    __restrict__ ws_keys,
                 const unsigned* __restrict__ ws_vals,
                 const int*      __restrict__ pair_idx,
                 int*            __restrict__ ws_sorted,
                 float*          __restrict__ out_pairs,
                 float*          __restrict__ out_labels)
{
    __shared__ float    sk[SORT_SIZE];
    __shared__ unsigned sv[SORT_SIZE];

    const int    b    = blockIdx.x;
    const int    tid  = threadIdx.x;
    const size_t base = (size_t)b * R_;

#if USE_ASYNC_LDS
    for (int t = tid; t < R_; t += SORT_NT) {
        __builtin_amdgcn_global_load_async_to_lds_b32(
            (int*)(ws_keys + base + t), (int*)&sk[t], 0, 0);
        __builtin_amdgcn_global_load_async_to_lds_b32(
            (int*)(ws_vals + base + t), (int*)&sv[t], 0, 0);
    }
    for (int t = R_ + tid; t < SORT_SIZE; t += SORT_NT) {
        sk[t] = -1.0f;            // keys are >= 0, so padding sinks to the end
        sv[t] = 0xFFFFFFFFu;
    }
    __builtin_amdgcn_s_wait_asynccnt(0);
    __syncthreads();
#else
    for (int t = tid; t < SORT_SIZE; t += SORT_NT) {
        if (t < R_) { sk[t] = ws_keys[base + t]; sv[t] = ws_vals[base + t]; }
        else        { sk[t] = -1.0f;             sv[t] = 0xFFFFFFFFu; }
    }
    __syncthreads();
#endif

    // Bitonic sort: "ascending" = best-first rank order.
    for (unsigned k = 2; k <= (unsigned)SORT_SIZE; k <<= 1) {
        for (unsigned j = k >> 1; j > 0; j >>= 1) {
            for (unsigned t = tid; t < SORT_SIZE / 2; t += SORT_NT) {
                const unsigned i = 2u * t - (t & (j - 1u));
                const unsigned p = i ^ j;
                const bool up = ((i & k) == 0u);
                const float    ka = sk[i], kb = sk[p];
                const unsigned va = sv[i], vb = sv[p];
                const bool b_before_a =
                    (kb > ka) || (kb == ka && (vb & 0xFFFFu) < (va & 0xFFFFu));
                const bool a_before_b =
                    (ka > kb) || (ka == kb && (va & 0xFFFFu) < (vb & 0xFFFFu));
                if (up ? b_before_a : a_before_b) {
                    sk[i] = kb; sk[p] = ka;
                    sv[i] = vb; sv[p] = va;
                }
            }
            __syncthreads();
        }
    }

    for (int t = tid; t < R_; t += SORT_NT) {
        const unsigned v = sv[t];
        const int r   = (int)(v & 0xFFFFu);
        const int cls = (int)(v >> 16);
        ws_sorted[base + t] = r;               // re-read by kernel 4: keep cached
        nt_store(&out_labels[base + t], (float)cls);
        const int2 pr = ((const int2*)pair_idx)[base + r];
        nt_store(&out_pairs[2 * (base + t) + 0], (float)pr.x);
        nt_store(&out_pairs[2 * (base + t) + 1], (float)pr.y);
    }
}

// ---------------------------------------------------------------------------
// Kernel 4: gathered softmax recompute -> rel_prob_sorted. One wave per output
// row; source rows should mostly hit L2 (165MB working set vs 192MB L2),
// protected by the NT output stores.
// ---------------------------------------------------------------------------
__global__ void rel_prob_kernel(const float* __restrict__ rel_logits,
                                const int*   __restrict__ ws_sorted,
                                float*       __restrict__ out_prob)
{
    const int wave = (int)((blockIdx.x * blockDim.x + threadIdx.x) >> 5);
    const int lane = threadIdx.x & 31;
    if (wave >= B_ * R_) return;

    const int b   = wave / R_;
    const int src = ws_sorted[wave];
    const float* base = rel_logits + ((size_t)b * R_ + src) * CR_;

    const float l0 = base[lane];
    const float l1 = (lane < CR_ - 32) ? base[lane + 32] : -FLT_MAX;

    float m = fmaxf(l0, l1);
    for (int o = 16; o > 0; o >>= 1) m = fmaxf(m, __shfl_xor(m, o, 32));

    const float e0 = __expf(l0 - m);
    const float e1 = (lane < CR_ - 32) ? __expf(l1 - m) : 0.f;
    float s = e0 + e1;
    for (int o = 16; o > 0; o >>= 1) s += __shfl_xor(s, o, 32);
    const float inv = 1.0f / s;

    float* op = out_prob + (size_t)wave * CR_;
    nt_store(&op[lane], e0 * inv);
    if (lane < CR_ - 32) nt_store(&op[lane + 32], e1 * inv);
}

// ---------------------------------------------------------------------------
extern "C" void kernel_launch(void* const* d_in, const int* in_sizes, int n_in,
                              void* d_out, int out_size, void* d_ws, size_t ws_size,
                              hipStream_t stream)
{
    const float* rel_logits = (const float*)d_in[0];
    const float* obj_logits = (const float*)d_in[1];
    const int*   pair_idx   = (const int*)d_in[2];
    float* out = (float*)d_out;

    // Workspace layout (~9.8 MB)
    float*    ws_obj    = (float*)d_ws;                               // B*N
    float*    ws_keys   = ws_obj + (size_t)B_ * N_;                   // B*R
    unsigned* ws_vals   = (unsigned*)(ws_keys + (size_t)B_ * R_);     // B*R
    int*      ws_sorted = (int*)(ws_vals + (size_t)B_ * R_);          // B*R

    // Output layout (flat, tuple order)
    const size_t OPRED = 0;
    const size_t OOSC  = OPRED + (size_t)B_ * N_;
    const size_t OPAIR = OOSC  + (size_t)B_ * N_;
    const size_t OPROB = OPAIR + (size_t)B_ * R_ * 2;
    const size_t OLBL  = OPROB + (size_t)B_ * R_ * CR_;

    {
        const long long threads = (long long)B_ * N_ * 32;
        obj_score_kernel<<<dim3((unsigned)((threads + 255) / 256)), dim3(256), 0, stream>>>(
            obj_logits, out + OPRED, out + OOSC, ws_obj);
    }
    {
        const long long threads = (long long)B_ * R_ * 32;
        rel_key_kernel<<<dim3((unsigned)((threads + 255) / 256)), dim3(256), 0, stream>>>(
            rel_logits, pair_idx, ws_obj, ws_keys, ws_vals);
    }
    sort_kernel<<<dim3(B_), dim3(SORT_NT), 0, stream>>>(
        ws_keys, ws_vals, pair_idx, ws_sorted, out + OPAIR, out + OLBL);
    {
        const long long threads = (long long)B_ * R_ * 32;
        rel_prob_kernel<<<dim3((unsigned)((threads + 255) / 256)), dim3(256), 0, stream>>>(
            rel_logits, ws_sorted, out + OPROB);
    }
}